// NonParametricMCCDOPD_15582141349977
// MI455X (gfx1250) — compile-verified
//
#include <hip/hip_runtime.h>
#include <hip/hip_bf16.h>

typedef __attribute__((ext_vector_type(2))) float v2f;
typedef __attribute__((ext_vector_type(8))) float v8f;

#define B_Q    2048
#define N_OBS  2048
#define P_POLY 15
#define G_GR   21
#define E_GR   64
#define K_TOT  36          // P_POLY + G_GR
#define D_OPD  256
#define NN     65536       // D*D
#define NKCH   9           // K_TOT / 4

// ---------------------------------------------------------------------------
// Kernel 1: brute-force 1-NN argmin (first-min tie-break like jnp.argmin)
// ---------------------------------------------------------------------------
__global__ __launch_bounds__(256) void nn_argmin_kernel(
    const float* __restrict__ pos, const float* __restrict__ obs,
    int* __restrict__ idx_out) {
  __shared__ float sx[256];
  __shared__ float sy[256];
  const int b = blockIdx.x * 256 + threadIdx.x;
  const float qx = pos[2 * b + 0];
  const float qy = pos[2 * b + 1];
  float best = 3.402823466e+38f;
  int besti = 0;
  for (int t = 0; t < N_OBS; t += 256) {
    const int j = t + threadIdx.x;
    sx[threadIdx.x] = obs[2 * j + 0];
    sy[threadIdx.x] = obs[2 * j + 1];
    __syncthreads();
    #pragma unroll 8
    for (int u = 0; u < 256; ++u) {
      const float dx = qx - sx[u];
      const float dy = qy - sy[u];
      const float d2 = dx * dx + dy * dy;
      if (d2 < best) { best = d2; besti = t + u; }   // strict < keeps first min
    }
    __syncthreads();
  }
  idx_out[b] = besti;
}

// ---------------------------------------------------------------------------
// Kernel 2: gather dictionary rows, mix with alphas -> A[b, 0:36]
//   A[b, 0:15]  = poly_dic[idx[b], :] @ alpha_poly   (15x15)
//   A[b, 15:36] = graph_dic[idx[b], :] @ alpha_graph (64x21)
// ---------------------------------------------------------------------------
__global__ __launch_bounds__(256) void coeff_kernel(
    const int* __restrict__ idx, const float* __restrict__ poly_dic,
    const float* __restrict__ graph_dic, const float* __restrict__ alpha_poly,
    const float* __restrict__ alpha_graph, float* __restrict__ A) {
  const int b = blockIdx.x * 256 + threadIdx.x;
  if (b >= B_Q) return;
  const int j = idx[b];
  const float* __restrict__ prow = poly_dic + (size_t)j * P_POLY;
  const float* __restrict__ grow = graph_dic + (size_t)j * E_GR;
  float* __restrict__ arow = A + (size_t)b * K_TOT;
  #pragma unroll
  for (int p = 0; p < P_POLY; ++p) {
    float acc = 0.f;
    #pragma unroll
    for (int q = 0; q < P_POLY; ++q) acc += prow[q] * alpha_poly[q * P_POLY + p];
    arow[p] = acc;
  }
  for (int g = 0; g < G_GR; ++g) {
    float acc = 0.f;
    #pragma unroll 8
    for (int e = 0; e < E_GR; ++e) acc += grow[e] * alpha_graph[e * G_GR + g];
    arow[P_POLY + g] = acc;
  }
}

// ---------------------------------------------------------------------------
// Kernel 3: opd[m, n] = sum_k A[m,k] * Scat[k,n]   (M=2048, K=36, N=65536)
// Scat rows 0..14 = S_poly[k, :, :] flat; rows 15..35 = S_graph[k-15, :, :].
// One wave -> 4 consecutive 16x16 output tiles; K=36 as 9x V_WMMA_F32_16X16X4.
// Block = 256 threads = 8 waves covering 16 rows x 512 cols of output.
// ---------------------------------------------------------------------------
__global__ __launch_bounds__(256) void opd_gemm_wmma_kernel(
    const float* __restrict__ A, const float* __restrict__ Sp,
    const float* __restrict__ Sg, float* __restrict__ out) {
  const int wave = threadIdx.x >> 5;
  const int lane = threadIdx.x & 31;
  const int half = lane >> 4;         // 0: lanes 0-15, 1: lanes 16-31
  const int l    = lane & 15;

  const int mtile = blockIdx.y;                    // 0..127
  const int ngrp  = blockIdx.x * 8 + wave;         // 0..1023 (each = 64 cols)
  const int n0    = ngrp * 64;
  const int m     = mtile * 16 + l;                // A row for this lane

  // --- A fragments: 16x4 f32 layout: vgpr0 = K{0|2}, vgpr1 = K{1|3} by half
  v2f afrag[NKCH];
  const float* __restrict__ arow = A + (size_t)m * K_TOT;
  #pragma unroll
  for (int c = 0; c < NKCH; ++c) {
    const int k0 = 4 * c + 2 * half;
    afrag[c].x = arow[k0];
    afrag[c].y = arow[k0 + 1];
  }

  v8f acc[4] = {};   // four 16x16 f32 accumulators (zero C)

  #pragma unroll
  for (int c = 0; c < NKCH; ++c) {
    const int kA = 4 * c + 2 * half;   // row feeding b.x (mirrors A layout)
    const int kB = kA + 1;             // row feeding b.y
    const float* __restrict__ rowA =
        (kA < P_POLY) ? (Sp + (size_t)kA * NN) : (Sg + (size_t)(kA - P_POLY) * NN);
    const float* __restrict__ rowB =
        (kB < P_POLY) ? (Sp + (size_t)kB * NN) : (Sg + (size_t)(kB - P_POLY) * NN);
    #pragma unroll
    for (int t = 0; t < 4; ++t) {
      const int n = n0 + t * 16 + l;
      v2f bfrag;
      bfrag.x = rowA[n];
      bfrag.y = rowB[n];
      // D = A(16x4) * B(4x16) + C ; 8 args: neg_a, A, neg_b, B, c_mod, C, reuse_a, reuse_b
      acc[t] = __builtin_amdgcn_wmma_f32_16x16x4_f32(
          false, afrag[c], false, bfrag, (short)0, acc[t], false, false);
    }
  }

  // --- Store: C/D layout vgpr r -> row (r + 8*half), col = l within tile.
  // Non-temporal: 512 MB write-once stream must not evict S from L2.
  const size_t rowbase = (size_t)(mtile * 16 + 8 * half) * NN;
  #pragma unroll
  for (int t = 0; t < 4; ++t) {
    float* __restrict__ col = out + rowbase + (size_t)(n0 + t * 16 + l);
    #pragma unroll
    for (int r = 0; r < 8; ++r) {
      __builtin_nontemporal_store(acc[t][r], col + (size_t)r * NN);
    }
  }
}

// ---------------------------------------------------------------------------
extern "C" void kernel_launch(void* const* d_in, const int* in_sizes, int n_in,
                              void* d_out, int out_size, void* d_ws, size_t ws_size,
                              hipStream_t stream) {
  const float* positions   = (const float*)d_in[0];  // [B, 2]
  const float* obs_pos     = (const float*)d_in[1];  // [N_OBS, 2]
  const float* poly_dic    = (const float*)d_in[2];  // [N_OBS, P]
  const float* graph_dic   = (const float*)d_in[3];  // [N_OBS, E]
  const float* alpha_poly  = (const float*)d_in[4];  // [P, P]
  const float* alpha_graph = (const float*)d_in[5];  // [E, G]
  const float* S_poly      = (const float*)d_in[6];  // [P, D, D]
  const float* S_graph     = (const float*)d_in[7];  // [G, D, D]
  float* out = (float*)d_out;                        // [B, D, D]

  int*   d_idx = (int*)d_ws;                                   // 2048 ints
  float* d_A   = (float*)((char*)d_ws + B_Q * sizeof(int));    // [B, 36] f32

  nn_argmin_kernel<<<dim3(B_Q / 256), dim3(256), 0, stream>>>(
      positions, obs_pos, d_idx);

  coeff_kernel<<<dim3(B_Q / 256), dim3(256), 0, stream>>>(
      d_idx, poly_dic, graph_dic, alpha_poly, alpha_graph, d_A);

  // Grid: x covers N in 512-col slabs (8 waves x 64 cols), y covers 128 M-tiles.
  opd_gemm_wmma_kernel<<<dim3(128, 128), dim3(256), 0, stream>>>(
      d_A, S_poly, S_graph, out);
}